// BasicDeconvolutionBlock_14285061226962
// MI455X (gfx1250) — compile-verified
//
#include <hip/hip_runtime.h>

typedef __attribute__((ext_vector_type(2))) float v2f;
typedef __attribute__((ext_vector_type(8))) float v8f;

#define N_IN   100000
#define N_OUT  250000
#define K3     27
#define CIN    64
#define COUT   64
#define TOT_OUT (N_OUT * COUT)          // 16,000,000 floats
#define TILES_PER_K (N_IN / 16)         // 6250 (exact)
#define WAVES_PER_BLOCK 8
#define BLOCKS_PER_K ((TILES_PER_K + WAVES_PER_BLOCK - 1) / WAVES_PER_BLOCK) // 782
#define BN_EPS 1e-5f
#define LEAKY 0.01f

// LDS layout for weight[k] (64x64 f32), arranged so the two consecutive
// K-values a lane needs sit in one b64 load, and the two lane-halves of a
// wave hit disjoint bank ranges (half0 -> banks 0..31, half1 -> banks 32..63):
//   idx(c,o) = (c>>2)*320 + ((c>>1)&1)*160 + (o<<1) + (c&1)
#define WLDS_SIZE (16 * 320)            // 5120 floats = 20 KB

// ---------------------------------------------------------------- zero pass
__global__ __launch_bounds__(256) void k_zero(float4* __restrict__ out,
                                              float* __restrict__ ws) {
    int gid = blockIdx.x * blockDim.x + threadIdx.x;
    int stride = gridDim.x * blockDim.x;
    for (int j = gid; j < TOT_OUT / 4; j += stride)
        out[j] = float4{0.f, 0.f, 0.f, 0.f};
    if (gid < 2 * COUT) ws[gid] = 0.f;   // [sum(64), sumsq(64)]
}

// ---------------------------------------------- WMMA gemm + scatter-add pass
__global__ __launch_bounds__(256) void k_gemm_scatter(
    const float* __restrict__ x, const float* __restrict__ w,
    const int* __restrict__ in_idx, const int* __restrict__ out_idx,
    float* __restrict__ out) {
    __shared__ float wlds[WLDS_SIZE];

    const int k = blockIdx.y;
    const float* wk = w + k * CIN * COUT;
    for (int i = threadIdx.x; i < CIN * COUT; i += blockDim.x) {
        int c = i >> 6, o = i & 63;
        wlds[(c >> 2) * 320 + ((c >> 1) & 1) * 160 + (o << 1) + (c & 1)] = wk[i];
    }
    __syncthreads();

    const int wave = threadIdx.x >> 5;
    const int lane = threadIdx.x & 31;
    const int tile = blockIdx.x * WAVES_PER_BLOCK + wave;
    if (tile >= TILES_PER_K) return;      // wave-uniform: EXEC stays all-1s

    const int p0    = tile << 4;
    const int half  = lane >> 4;          // 0: lanes 0-15, 1: lanes 16-31
    const int lncol = lane & 15;

    // A fragment source row for this lane (ISA 32-bit 16x4 A layout:
    // lane<16 holds M=lane,K=4c..4c+1 ; lane>=16 holds M=lane-16,K=4c+2..4c+3)
    const int srow = in_idx[k * N_IN + p0 + lncol];
    const float* xrow = x + srow * CIN + (half << 1);

    v8f acc0 = {}, acc1 = {}, acc2 = {}, acc3 = {};

#pragma unroll
    for (int c0 = 0; c0 < CIN / 4; ++c0) {
        v2f a = *(const v2f*)(xrow + (c0 << 2));
        const float* bb = wlds + c0 * 320 + half * 160 + (lncol << 1);
        v2f b0 = *(const v2f*)(bb);
        v2f b1 = *(const v2f*)(bb + 32);
        v2f b2 = *(const v2f*)(bb + 64);
        v2f b3 = *(const v2f*)(bb + 96);
        acc0 = __builtin_amdgcn_wmma_f32_16x16x4_f32(false, a, false, b0,
                                                     (short)0, acc0, false, false);
        acc1 = __builtin_amdgcn_wmma_f32_16x16x4_f32(false, a, false, b1,
                                                     (short)0, acc1, false, false);
        acc2 = __builtin_amdgcn_wmma_f32_16x16x4_f32(false, a, false, b2,
                                                     (short)0, acc2, false, false);
        acc3 = __builtin_amdgcn_wmma_f32_16x16x4_f32(false, a, false, b3,
                                                     (short)0, acc3, false, false);
    }

    // C/D layout: VGPR r, lane<16 -> M=r, N=lane ; lane>=16 -> M=8+r, N=lane-16
    const int obase = k * N_IN + p0 + (half << 3);
#pragma unroll
    for (int r = 0; r < 8; ++r) {
        const int q = out_idx[obase + r];
        float* orow = out + q * COUT + lncol;
        atomicAdd(orow + 0,  acc0[r]);
        atomicAdd(orow + 16, acc1[r]);
        atomicAdd(orow + 32, acc2[r]);
        atomicAdd(orow + 48, acc3[r]);
    }
}

// ------------------------------------------------------- per-channel stats
__global__ __launch_bounds__(256) void k_stats(const float* __restrict__ out,
                                               float* __restrict__ ws) {
    __shared__ float ssum[COUT], ssq[COUT];
    int t = threadIdx.x;
    if (t < COUT) { ssum[t] = 0.f; ssq[t] = 0.f; }
    __syncthreads();

    int gid = blockIdx.x * blockDim.x + t;
    int stride = gridDim.x * blockDim.x;   // multiple of 64 -> channel fixed
    int c = gid & 63;
    float s = 0.f, q = 0.f;
    for (int i = gid; i < TOT_OUT; i += stride) {
        float v = out[i];
        s += v;
        q += v * v;
    }
    atomicAdd(&ssum[c], s);
    atomicAdd(&ssq[c], q);
    __syncthreads();
    if (t < COUT) {
        atomicAdd(&ws[t], ssum[t]);
        atomicAdd(&ws[COUT + t], ssq[t]);
    }
}

// ---------------------------------------------- BN + LeakyReLU (in place)
__global__ __launch_bounds__(256) void k_finalize(float* __restrict__ out,
                                                  const float* __restrict__ ws,
                                                  const float* __restrict__ gamma,
                                                  const float* __restrict__ beta) {
    __shared__ float sc[COUT], sh[COUT];
    int t = threadIdx.x;
    if (t < COUT) {
        float mean = ws[t] * (1.0f / N_OUT);
        float var  = ws[COUT + t] * (1.0f / N_OUT) - mean * mean;
        float rs   = rsqrtf(var + BN_EPS);
        float g    = gamma[t];
        sc[t] = rs * g;
        sh[t] = beta[t] - mean * rs * g;
    }
    __syncthreads();

    int gid = blockIdx.x * blockDim.x + t;
    int stride = gridDim.x * blockDim.x;
    for (int i = gid; i < TOT_OUT; i += stride) {
        int c = i & 63;
        float v = out[i] * sc[c] + sh[c];
        out[i] = (v >= 0.f) ? v : LEAKY * v;
    }
}

extern "C" void kernel_launch(void* const* d_in, const int* in_sizes, int n_in,
                              void* d_out, int out_size, void* d_ws, size_t ws_size,
                              hipStream_t stream) {
    const float* x       = (const float*)d_in[0];
    const float* weight  = (const float*)d_in[1];
    const float* gamma   = (const float*)d_in[2];
    const float* beta    = (const float*)d_in[3];
    const int*   in_idx  = (const int*)d_in[4];
    const int*   out_idx = (const int*)d_in[5];
    float* out = (float*)d_out;
    float* ws  = (float*)d_ws;

    k_zero<<<2048, 256, 0, stream>>>((float4*)out, ws);

    dim3 grid(BLOCKS_PER_K, K3);
    k_gemm_scatter<<<grid, 256, 0, stream>>>(x, weight, in_idx, out_idx, out);

    k_stats<<<1024, 256, 0, stream>>>(out, ws);
    k_finalize<<<2048, 256, 0, stream>>>(out, ws, gamma, beta);
}